// TripletLoss_72825465471198
// MI455X (gfx1250) — compile-verified
//
#include <hip/hip_runtime.h>

#define NROWS   4096
#define DIM     512
#define MARGIN_F 0.2f
#define JT      128     // columns per block (4 quarters x 32 cols)
#define ICH     512     // rows per block (I-chunk)

typedef __attribute__((ext_vector_type(2))) float v2f;
typedef __attribute__((ext_vector_type(8))) float v8f;

// ---------------------------------------------------------------------------
// Kernel 1: per-row inverse L2 norms, normalized diagonal score, and zero-init
// of the per-column max buffer (graph-replay safe). One wave per row.
// ---------------------------------------------------------------------------
__global__ __launch_bounds__(256) void prep_kernel(const float* __restrict__ brand,
                                                   const float* __restrict__ post,
                                                   float* __restrict__ invb,
                                                   float* __restrict__ invp,
                                                   float* __restrict__ diag,
                                                   float* __restrict__ colmax)
{
    int row  = blockIdx.x * 8 + (threadIdx.x >> 5);
    int lane = threadIdx.x & 31;
    const float* br = brand + (size_t)row * DIM;
    const float* pr = post  + (size_t)row * DIM;
    float sb = 0.f, sp = 0.f, sd = 0.f;
#pragma unroll 4
    for (int k = lane; k < DIM; k += 32) {
        float b = br[k];
        float p = pr[k];
        sb = fmaf(b, b, sb);
        sp = fmaf(p, p, sp);
        sd = fmaf(b, p, sd);
    }
#pragma unroll
    for (int off = 16; off > 0; off >>= 1) {
        sb += __shfl_xor(sb, off, 32);
        sp += __shfl_xor(sp, off, 32);
        sd += __shfl_xor(sd, off, 32);
    }
    if (lane == 0) {
        float ib = rsqrtf(sb);
        float ip = rsqrtf(sp);
        invb[row]   = ib;
        invp[row]   = ip;
        diag[row]   = sd * ib * ip;   // normalized s[j,j]
        colmax[row] = 0.0f;
    }
}

// ---------------------------------------------------------------------------
// Kernel 2: fused GEMM (v_wmma_f32_16x16x4_f32) + margin/mask + column max.
// Block = 256 threads = 8 waves. Each wave computes a 32x32 C tile as a 2x2
// grid of 16x16 WMMA fragments: 4 WMMA per 4 global b64 loads (1:1 ratio).
// Wave w: row-half (w>>2) of the 64-row step, column-quarter (w&3) of the
// 128-col block tile. Normalization applied to the accumulator post-K-loop.
// ---------------------------------------------------------------------------
__global__ __launch_bounds__(256) void score_kernel(const float* __restrict__ brand,
                                                    const float* __restrict__ post,
                                                    const int*   __restrict__ ids,
                                                    const float* __restrict__ invb,
                                                    const float* __restrict__ invp,
                                                    const float* __restrict__ diag,
                                                    float*       __restrict__ colmax)
{
    const int njt = NROWS / JT;              // 32 column tiles
    int jt = blockIdx.x % njt;
    int ic = blockIdx.x / njt;
    int j_base = jt * JT;
    int i_base = ic * ICH;

    int tid    = threadIdx.x;
    int wave   = tid >> 5;
    int lane   = tid & 31;
    int lane16 = lane & 15;
    int hi     = lane >> 4;                  // 0: K pair {0,1} / M 0..7, 1: {2,3} / M 8..15

    int cq = wave & 3;                       // column quarter: 32 cols
    int rh = wave >> 1 >> 1;                 // row half: 32 rows of each 64-row step

    __shared__ int   s_ids[ICH];
    __shared__ float s_invb[ICH];
    for (int t = tid; t < ICH; t += 256) {
        s_ids[t]  = ids[i_base + t];
        s_invb[t] = invb[i_base + t];
    }
    __syncthreads();

    int   j0   = j_base + cq * 32 + lane16;  // this lane's two columns
    int   j1   = j0 + 16;
    float dj0  = diag[j0],  dj1  = diag[j1];
    float ipj0 = invp[j0],  ipj1 = invp[j1];
    int   idj0 = ids[j0],   idj1 = ids[j1];

    const float* bp0 = post + (size_t)j0 * DIM + 2 * hi;
    const float* bp1 = post + (size_t)j1 * DIM + 2 * hi;
    float cmax0 = 0.0f, cmax1 = 0.0f;

    for (int it = 0; it < ICH / 64; ++it) {
        int i0 = i_base + it * 64 + rh * 32;
        const float* ap0 = brand + (size_t)(i0 + lane16) * DIM + 2 * hi;
        const float* ap1 = ap0 + 16 * (size_t)DIM;
        if (it + 1 < ICH / 64) {             // prefetch next step's A rows
            __builtin_prefetch(ap0 + 64 * (size_t)DIM, 0, 0);
            __builtin_prefetch(ap1 + 64 * (size_t)DIM, 0, 0);
        }
        v8f acc00 = {}, acc01 = {}, acc10 = {}, acc11 = {};
#pragma unroll 4
        for (int k = 0; k < DIM; k += 4) {
            v2f a0 = *(const v2f*)(ap0 + k);
            v2f a1 = *(const v2f*)(ap1 + k);
            v2f b0 = *(const v2f*)(bp0 + k);
            v2f b1 = *(const v2f*)(bp1 + k);
            acc00 = __builtin_amdgcn_wmma_f32_16x16x4_f32(false, a0, false, b0,
                                                          (short)0, acc00, false, false);
            acc01 = __builtin_amdgcn_wmma_f32_16x16x4_f32(false, a0, false, b1,
                                                          (short)0, acc01, false, false);
            acc10 = __builtin_amdgcn_wmma_f32_16x16x4_f32(false, a1, false, b0,
                                                          (short)0, acc10, false, false);
            acc11 = __builtin_amdgcn_wmma_f32_16x16x4_f32(false, a1, false, b1,
                                                          (short)0, acc11, false, false);
        }
        int mb = it * 64 + rh * 32 + hi * 8;
#pragma unroll
        for (int v = 0; v < 8; ++v) {
            int   m0  = mb + v;
            int   m1  = m0 + 16;
            float ib0 = s_invb[m0], ib1 = s_invb[m1];
            int   id0 = s_ids[m0],  id1 = s_ids[m1];
            float c;
            c = fmaxf(acc00[v] * ib0 * ipj0 + MARGIN_F - dj0, 0.0f);
            if (id0 == idj0) c = 0.0f;
            cmax0 = fmaxf(cmax0, c);
            c = fmaxf(acc10[v] * ib1 * ipj0 + MARGIN_F - dj0, 0.0f);
            if (id1 == idj0) c = 0.0f;
            cmax0 = fmaxf(cmax0, c);
            c = fmaxf(acc01[v] * ib0 * ipj1 + MARGIN_F - dj1, 0.0f);
            if (id0 == idj1) c = 0.0f;
            cmax1 = fmaxf(cmax1, c);
            c = fmaxf(acc11[v] * ib1 * ipj1 + MARGIN_F - dj1, 0.0f);
            if (id1 == idj1) c = 0.0f;
            cmax1 = fmaxf(cmax1, c);
        }
    }

    // lanes L and L^16 hold the same column: merge halves (wave32 shuffle)
    cmax0 = fmaxf(cmax0, __shfl_xor(cmax0, 16, 32));
    cmax1 = fmaxf(cmax1, __shfl_xor(cmax1, 16, 32));
    if (hi == 0) {
        // costs are >= 0, so integer max on the float bit pattern is exact
        atomicMax((int*)(colmax + j0), __float_as_int(cmax0));
        atomicMax((int*)(colmax + j1), __float_as_int(cmax1));
    }
}

// ---------------------------------------------------------------------------
// Kernel 3: sum the 4096 column maxima -> scalar output (single block).
// ---------------------------------------------------------------------------
__global__ __launch_bounds__(256) void reduce_kernel(const float* __restrict__ colmax,
                                                     float* __restrict__ out)
{
    __shared__ float s[256];
    float v = 0.0f;
    for (int i = threadIdx.x; i < NROWS; i += 256) v += colmax[i];
    s[threadIdx.x] = v;
    __syncthreads();
    for (int off = 128; off > 0; off >>= 1) {
        if (threadIdx.x < off) s[threadIdx.x] += s[threadIdx.x + off];
        __syncthreads();
    }
    if (threadIdx.x == 0) out[0] = s[0];
}

extern "C" void kernel_launch(void* const* d_in, const int* in_sizes, int n_in,
                              void* d_out, int out_size, void* d_ws, size_t ws_size,
                              hipStream_t stream) {
    const float* brand = (const float*)d_in[0];
    const float* post  = (const float*)d_in[1];
    const int*   ids   = (const int*)d_in[2];
    float*       out   = (float*)d_out;

    float* ws     = (float*)d_ws;           // 64 KB total
    float* invb   = ws;
    float* invp   = ws + NROWS;
    float* diag   = ws + 2 * NROWS;
    float* colmax = ws + 3 * NROWS;

    prep_kernel<<<NROWS / 8, 256, 0, stream>>>(brand, post, invb, invp, diag, colmax);

    int grid = (NROWS / ICH) * (NROWS / JT);  // 8 * 32 = 256 blocks
    score_kernel<<<grid, 256, 0, stream>>>(brand, post, ids, invb, invp, diag, colmax);

    reduce_kernel<<<1, 256, 0, stream>>>(colmax, out);
}